// Node2Edge_64948495450698
// MI455X (gfx1250) — compile-verified
//
#include <hip/hip_runtime.h>

// Shapes from the reference
#define Bq 4
#define Cq 256
#define Eq 32
#define Nq 1024

typedef float v2f __attribute__((ext_vector_type(2)));
typedef float v4f __attribute__((ext_vector_type(4)));
typedef float v8f __attribute__((ext_vector_type(8)));

// ---------------------------------------------------------------------------
// Kernel 1: t12[b][e][n] = sum_c theta12[e][c] * emb[b][c][n]
// One wave (32 threads) per 16x16 output tile, fp32 WMMA 16x16x4, K-loop of 64.
// Grid: Bq * (Eq/16) * (Nq/16) = 4*2*64 = 512 blocks of 32 threads.
// ---------------------------------------------------------------------------
__global__ __launch_bounds__(32)
void t12_wmma_kernel(const float* __restrict__ theta12,
                     const float* __restrict__ emb,
                     float* __restrict__ t12) {
    const int bid   = blockIdx.x;
    const int ntile = bid & 63;            // 64 N-tiles
    const int etile = (bid >> 6) & 1;      // 2 E-tiles
    const int b     = bid >> 7;            // batch
    const int lane  = threadIdx.x;         // 0..31
    const int half  = lane >> 4;           // 0: K=0,1   1: K=2,3
    const int lid   = lane & 15;           // M index (A) / N index (B)

    // A: theta12 row for m = etile*16 + lid  (16x4 tile per step)
    const float* __restrict__ Arow = theta12 + (size_t)(etile * 16 + lid) * Cq;
    // B: emb column n = ntile*16 + lid of batch b (4x16 tile per step)
    const float* __restrict__ Bcol = emb + (size_t)b * Cq * Nq + ntile * 16 + lid;

    v8f acc = {};
    #pragma unroll 4
    for (int k0 = 0; k0 < Cq; k0 += 4) {
        // A 16x4 fp32 layout: VGPR0 = K=(2*half), VGPR1 = K=(2*half+1)
        v2f a = *(const v2f*)(Arow + k0 + 2 * half);
        // B 4x16 fp32 layout mirrored: lanes 0-15 hold K=0,1; lanes 16-31 K=2,3
        v2f bm;
        bm.x = Bcol[(size_t)(k0 + 2 * half + 0) * Nq];
        bm.y = Bcol[(size_t)(k0 + 2 * half + 1) * Nq];
        acc = __builtin_amdgcn_wmma_f32_16x16x4_f32(
            /*neg_a=*/false, a, /*neg_b=*/false, bm,
            /*c_mod=*/(short)0, acc, /*reuse_a=*/false, /*reuse_b=*/false);
    }

    // D layout: VGPR m -> row (half*8 + m), lane lid -> column
    float* __restrict__ Orow =
        t12 + ((size_t)b * Eq + etile * 16 + half * 8) * Nq + ntile * 16 + lid;
    #pragma unroll
    for (int m = 0; m < 8; ++m)
        Orow[(size_t)m * Nq] = acc[m];
}

// ---------------------------------------------------------------------------
// Kernel 2: out[b,e,i,j] = softmax_j( 2*max(t[i],t[j]) + theta5[e]*(i==j) )
// Grid: (Bq*Eq)*8 = 1024 blocks, 256 threads (8 waves). Block loads t (1024 f)
// into LDS once, each wave handles one row at a time (wave-shuffle reductions),
// 16 rows per wave -> 128 rows per block. Output stores are NT b128.
// ---------------------------------------------------------------------------
__global__ __launch_bounds__(256)
void softmax_rows_kernel(const float* __restrict__ t12,
                         const float* __restrict__ theta5,
                         float* __restrict__ out) {
    __shared__ float sm[Nq];

    const int be    = blockIdx.x >> 3;     // 0..127  (b*Eq + e)
    const int chunk = blockIdx.x & 7;      // row chunk of 128
    const int lane  = threadIdx.x & 31;
    const int wave  = threadIdx.x >> 5;    // 0..7
    const int e     = be & (Eq - 1);

    // Cooperative LDS fill of the t-vector for this (b,e): 256 x float4
    ((v4f*)sm)[threadIdx.x] = ((const v4f*)(t12 + (size_t)be * Nq))[threadIdx.x];
    __syncthreads();

    const float th5 = theta5[e];

    #pragma unroll 1
    for (int it = 0; it < 16; ++it) {
        const int i  = chunk * 128 + it * 8 + wave;
        const float ti = sm[i];

        // Pass 1: values + row max (lane covers j = g*128 + lane*4 + c)
        v4f  vals[8];
        float m = -3.402823466e38f;
        #pragma unroll
        for (int g = 0; g < 8; ++g) {
            const int j0 = g * 128 + lane * 4;
            v4f tj = *(const v4f*)(sm + j0);
            v4f v;
            #pragma unroll
            for (int c = 0; c < 4; ++c) {
                float vv = 2.0f * fmaxf(ti, tj[c]);
                if (j0 + c == i) vv += th5;       // exact diagonal term
                v[c] = vv;
                m = fmaxf(m, vv);
            }
            vals[g] = v;
        }
        #pragma unroll
        for (int off = 16; off > 0; off >>= 1)
            m = fmaxf(m, __shfl_xor(m, off, 32));

        // Pass 2: exp + row sum
        float s = 0.0f;
        #pragma unroll
        for (int g = 0; g < 8; ++g) {
            v4f v = vals[g];
            #pragma unroll
            for (int c = 0; c < 4; ++c) {
                const float ex = __expf(v[c] - m);
                v[c] = ex;
                s += ex;
            }
            vals[g] = v;
        }
        #pragma unroll
        for (int off = 16; off > 0; off >>= 1)
            s += __shfl_xor(s, off, 32);
        const float inv = __builtin_amdgcn_rcpf(s);

        // Pass 3: normalize + coalesced NT stores (b128 per lane)
        float* __restrict__ orow = out + ((size_t)be * Nq + i) * Nq;
        #pragma unroll
        for (int g = 0; g < 8; ++g) {
            v4f v = vals[g] * inv;
            __builtin_nontemporal_store(v, (v4f*)(orow + g * 128 + lane * 4));
        }
    }
}

// ---------------------------------------------------------------------------
extern "C" void kernel_launch(void* const* d_in, const int* in_sizes, int n_in,
                              void* d_out, int out_size, void* d_ws, size_t ws_size,
                              hipStream_t stream) {
    const float* emb     = (const float*)d_in[0];  // (B,C,N)
    const float* theta12 = (const float*)d_in[1];  // (1,E,C)
    // d_in[2] = theta34 : unused by the reference
    const float* theta5  = (const float*)d_in[3];  // (1,E,1,1) -> E floats
    float*       out     = (float*)d_out;          // (B,E,N,N)
    float*       t12     = (float*)d_ws;           // B*E*N floats = 512 KB scratch

    t12_wmma_kernel<<<Bq * (Eq / 16) * (Nq / 16), 32, 0, stream>>>(theta12, emb, t12);
    softmax_rows_kernel<<<(Bq * Eq) * 8, 256, 0, stream>>>(t12, theta5, out);
}